// soft_margin_softmax_40089224740959
// MI455X (gfx1250) — compile-verified
//
#include <hip/hip_runtime.h>

// Soft-margin softmax over [4096, 32000] f32.
//   out = s*e / (S + (s-1)*e),  e = exp(x), s = exp(-dist), S = row sum of e.
// Single-pass, memory-bound design for MI455X (23.3 TB/s HBM):
//   - one workgroup per row, 800 threads (25 wave32 waves), 10 float4/thread
//   - exp(x) kept resident in VGPRs between the reduction and the store
//   - non-temporal 128-bit loads/stores (streaming working set >> 192MB L2)
//   - wave-level reduction via V_WMMA_F32_16X16X4_F32 (B = ones => D rows are
//     A row-sums; summing the 8 D VGPRs + one xor-16 shuffle yields the exact
//     sum of all A entries, independent of A's internal lane layout)

typedef __attribute__((ext_vector_type(4))) float f4;
typedef __attribute__((ext_vector_type(2))) float v2f;
typedef __attribute__((ext_vector_type(8))) float v8f;

#define COLS   32000
#define BLOCK  800           // 25 wave32 waves
#define NF4    10            // float4 per thread: 800*10*4 == 32000
#define NWAVES (BLOCK / 32)

__global__ __launch_bounds__(BLOCK) void soft_margin_softmax_kernel(
    const float* __restrict__ logits,
    const float* __restrict__ dist_p,
    float* __restrict__ out)
{
    const int row = blockIdx.x;
    const int tid = threadIdx.x;
    const size_t rowoff = (size_t)row * (size_t)COLS;
    const f4* __restrict__ in4  = (const f4*)(logits + rowoff);
    f4*       __restrict__ out4 = (f4*)(out + rowoff);

    const float s = __expf(-dist_p[0]);   // exp(-dist), scalar broadcast

    // ---- Phase 1: stream the row once, keep e = exp(x) in registers ----
    f4 ev[NF4];
    float p = 0.0f;
#pragma unroll
    for (int i = 0; i < NF4; ++i) {
        f4 x = __builtin_nontemporal_load(&in4[i * BLOCK + tid]);
        f4 e;
        e.x = __expf(x.x);
        e.y = __expf(x.y);
        e.z = __expf(x.z);
        e.w = __expf(x.w);
        ev[i] = e;
        p += (e.x + e.y) + (e.z + e.w);
    }

    // ---- Wave reduction of p via V_WMMA_F32_16X16X4_F32 ----
    // A: VGPR0 = p, VGPR1 = 0 ; B = all ones ; C = 0.
    // D[m][n] = rowsum_m(A). Sum of d0..d7 per lane covers M=0..7 (lanes 0-15)
    // or M=8..15 (lanes 16-31) at fixed N; pairing lane L with L^16 covers all
    // M, i.e. the sum of every A element == sum of p across the wave.
    v2f a; a.x = p;    a.y = 0.0f;
    v2f b; b.x = 1.0f; b.y = 1.0f;
    v8f c = {};
    v8f d = __builtin_amdgcn_wmma_f32_16x16x4_f32(
        /*neg_a=*/false, a, /*neg_b=*/false, b,
        /*c_mod=*/(short)0, c, /*reuse_a=*/false, /*reuse_b=*/false);
    float r = ((d[0] + d[1]) + (d[2] + d[3])) + ((d[4] + d[5]) + (d[6] + d[7]));
    float wavesum = r + __shfl_xor(r, 16, 32);

    // ---- Cross-wave combine through LDS (25 partials) ----
    __shared__ float wsum[NWAVES];
    const int wid = tid >> 5;
    if ((tid & 31) == 0) wsum[wid] = wavesum;
    __syncthreads();

    float S = 0.0f;
#pragma unroll
    for (int w = 0; w < NWAVES; ++w) S += wsum[w];

    // ---- Phase 2: out = s*e / (S + (s-1)*e), streamed store ----
    const float sm1 = s - 1.0f;
#pragma unroll
    for (int i = 0; i < NF4; ++i) {
        f4 e = ev[i];
        f4 o;
        o.x = (s * e.x) / (S + sm1 * e.x);
        o.y = (s * e.y) / (S + sm1 * e.y);
        o.z = (s * e.z) / (S + sm1 * e.z);
        o.w = (s * e.w) / (S + sm1 * e.w);
        __builtin_nontemporal_store(o, &out4[i * BLOCK + tid]);
    }
}

extern "C" void kernel_launch(void* const* d_in, const int* in_sizes, int n_in,
                              void* d_out, int out_size, void* d_ws, size_t ws_size,
                              hipStream_t stream) {
    const float* logits = (const float*)d_in[0];
    const float* dist   = (const float*)d_in[1];
    float* out = (float*)d_out;

    const int rows = out_size / COLS;   // 4096
    dim3 grid(rows), block(BLOCK);
    hipLaunchKernelGGL(soft_margin_softmax_kernel, grid, block, 0, stream,
                       logits, dist, out);
}